// ts_up_sampling_28054726378002
// MI455X (gfx1250) — compile-verified
//
#include <hip/hip_runtime.h>
#include <hip/hip_bf16.h>

// ---------------------------------------------------------------------------
// DGCNN-style pipeline for MI455X (gfx1250, wave32, WMMA).
// Big GEMMs (pairwise Gram, edge-conv K=128, head 256x192) use
// V_WMMA_F32_16X16X4_F32 so neighbor ordering stays exact fp32.
// pd (128 MB) is never materialized: Gram tiles + running top-4 in registers.
// A transposed feature copy ht[n][64] makes every WMMA operand fetch a
// contiguous b64; each wave processes TWO query tiles per candidate fetch
// (16 loads : 32 wmma per tile).
// ---------------------------------------------------------------------------

typedef __attribute__((ext_vector_type(2))) float v2f;
typedef __attribute__((ext_vector_type(8))) float v8f;

#define NBATCH 8
#define NPTS   2048

__device__ __forceinline__ v8f wmma_f32(v2f a, v2f b, v8f c) {
  // D(16x16) = A(16x4) * B(4x16) + C, fp32
  return __builtin_amdgcn_wmma_f32_16x16x4_f32(
      /*neg_a=*/false, a, /*neg_b=*/false, b,
      /*c_mod=*/(short)0, c, /*reuse_a=*/false, /*reuse_b=*/false);
}

struct Top4 {
  float v0, v1, v2, v3;
  int   i0, i1, i2, i3;
  __device__ __forceinline__ void init() {
    v0 = v1 = v2 = v3 = -1e30f;
    i0 = i1 = i2 = i3 = 0;
  }
  // streaming insert: strict '>' so earlier (lower) index wins ties (JAX order)
  __device__ __forceinline__ void insert(float k, int c) {
    if (k > v3) {
      if (k > v0)      { v3=v2;i3=i2; v2=v1;i2=i1; v1=v0;i1=i0; v0=k;i0=c; }
      else if (k > v1) { v3=v2;i3=i2; v2=v1;i2=i1; v1=k;i1=c; }
      else if (k > v2) { v3=v2;i3=i2; v2=k;i2=c; }
      else             { v3=k;i3=c; }
    }
  }
  __device__ __forceinline__ static bool beats(float nv, int ni, float v, int i) {
    return (nv > v) || (nv == v && ni < i);
  }
  // merge insert with explicit tie-break on index
  __device__ __forceinline__ void insert_tb(float nv, int ni) {
    if (beats(nv,ni,v3,i3)) {
      if (beats(nv,ni,v0,i0))      { v3=v2;i3=i2; v2=v1;i2=i1; v1=v0;i1=i0; v0=nv;i0=ni; }
      else if (beats(nv,ni,v1,i1)) { v3=v2;i3=i2; v2=v1;i2=i1; v1=nv;i1=ni; }
      else if (beats(nv,ni,v2,i2)) { v3=v2;i3=i2; v2=nv;i2=ni; }
      else                         { v3=nv;i3=ni; }
    }
  }
  // fold lane+16's list into lanes 0..15
  __device__ __forceinline__ void merge_partner(int lane) {
    float a0 = __shfl(v0, lane + 16, 32), a1 = __shfl(v1, lane + 16, 32);
    float a2 = __shfl(v2, lane + 16, 32), a3 = __shfl(v3, lane + 16, 32);
    int   b0 = __shfl(i0, lane + 16, 32), b1 = __shfl(i1, lane + 16, 32);
    int   b2 = __shfl(i2, lane + 16, 32), b3 = __shfl(i3, lane + 16, 32);
    if (lane < 16) {
      insert_tb(a0, b0); insert_tb(a1, b1); insert_tb(a2, b2); insert_tb(a3, b3);
    }
  }
};

// ---------------------------------------------------------------------------
// K1: fused conv_dup (256x128) -> reshape(256,1024 -> 128,2048) -> conv_c1
//     (64x128). Writes h[8][64][2048] (channel-major, for the g-max) AND
//     ht[8][2048][64] (point-major, for all WMMA consumers).
// reshape: hd2[c2][n2] = hd[2*c2 + (n2>=1024)][n2 % 1024]
// ---------------------------------------------------------------------------
__global__ __launch_bounds__(128) void k_front(
    const float* __restrict__ x,
    const float* __restrict__ Wd, const float* __restrict__ sd, const float* __restrict__ bd,
    const float* __restrict__ Wc, const float* __restrict__ sc, const float* __restrict__ bc,
    float* __restrict__ h, float* __restrict__ ht) {
  __shared__ float xs[128 * 32];   // x tile   (16 KB)
  __shared__ float hds[256 * 32];  // hd tile  (32 KB)
  const int b   = blockIdx.x;
  const int nb0 = blockIdx.y * 32;
  const int t   = threadIdx.x;
  const float* xb = x + (size_t)b * 128 * 1024;

  for (int r = 0; r < 32; ++r) {
    int c = r * 4 + (t >> 5);
    int j = t & 31;
    xs[c * 32 + j] = xb[c * 1024 + nb0 + j];
  }
  __syncthreads();

  for (int i = 0; i < 64; ++i) {
    int id = i * 128 + t;
    int c = id >> 5;     // uniform per wave -> scalar weight loads
    int j = id & 31;
    float acc = 0.f;
    const float* w = Wd + c * 128;
#pragma unroll 8
    for (int k = 0; k < 128; ++k) acc = fmaf(w[k], xs[k * 32 + j], acc);
    acc = acc * sd[c] + bd[c];
    hds[c * 32 + j] = acc > 0.f ? acc : 0.f;
  }
  __syncthreads();

  float* hb  = h  + (size_t)b * 64 * 2048;
  float* hbt = ht + (size_t)b * 2048 * 64;
  for (int i = 0; i < 32; ++i) {
    int id  = i * 128 + t;
    int o   = id >> 6;
    int col = id & 63;
    int hi  = col >> 5;
    int j   = col & 31;
    float acc = 0.f;
    const float* w = Wc + o * 128;
#pragma unroll 8
    for (int c2 = 0; c2 < 128; ++c2)
      acc = fmaf(w[c2], hds[(2 * c2 + hi) * 32 + j], acc);
    acc = acc * sc[o] + bc[o];
    float r = acc > 0.f ? acc : 0.f;
    int n2 = nb0 + j + hi * 1024;
    hb[o * 2048 + n2] = r;
    hbt[(size_t)n2 * 64 + o] = r;
  }
}

// ---------------------------------------------------------------------------
// K2: per-batch stats: xx[n] = |h_n|^2 (contiguous from ht), g[c] = max_n h,
//     then the tiny global MLP -> gfin[8][128]
// ---------------------------------------------------------------------------
__global__ __launch_bounds__(256) void k_stats(
    const float* __restrict__ h, const float* __restrict__ ht,
    const float* __restrict__ W3a, const float* __restrict__ s3a, const float* __restrict__ b3a,
    const float* __restrict__ W3b, const float* __restrict__ s3b, const float* __restrict__ b3b,
    float* __restrict__ xx, float* __restrict__ gfin) {
  __shared__ float gs[64];
  __shared__ float g1[128];
  const int b = blockIdx.x;
  const int t = threadIdx.x;
  const float* hb  = h  + (size_t)b * 64 * 2048;
  const float* hbt = ht + (size_t)b * 2048 * 64;

  for (int n = t; n < 2048; n += 256) {
    const float* p = hbt + (size_t)n * 64;
    float s = 0.f;
#pragma unroll 8
    for (int c = 0; c < 64; ++c) { float v = p[c]; s = fmaf(v, v, s); }
    xx[b * 2048 + n] = s;
  }
  if (t < 64) {
    float m = -1e30f;
    for (int n = 0; n < 2048; ++n) m = fmaxf(m, hb[t * 2048 + n]);
    gs[t] = m;
  }
  __syncthreads();
  if (t < 128) {
    float acc = 0.f;
    const float* w = W3a + t * 64;
    for (int c = 0; c < 64; ++c) acc = fmaf(w[c], gs[c], acc);
    acc = acc * s3a[t] + b3a[t];
    g1[t] = acc > 0.f ? acc : 0.f;
  }
  __syncthreads();
  if (t < 128) {
    float acc = 0.f;
    const float* w = W3b + t * 128;
    for (int c = 0; c < 128; ++c) acc = fmaf(w[c], g1[c], acc);
    acc = acc * s3b[t] + b3b[t];
    gfin[b * 128 + t] = acc > 0.f ? acc : 0.f;
  }
}

// ---------------------------------------------------------------------------
// K3: fused pairwise Gram (WMMA f32 16x16x4) + running top-4.
// Wave = 32 queries (two 16-query tiles sharing each candidate fetch).
// D[M=cand][N=query]: lane (half,m) accumulates 8 candidates for query m.
// Rank key = inner - 0.5*xx[cand] (monotone-equal to pd per query).
// ---------------------------------------------------------------------------
__global__ __launch_bounds__(128) void k_knn(
    const float* __restrict__ ht, const float* __restrict__ xx,
    int* __restrict__ nidx) {
  __shared__ float xs[2048];
  const int b = blockIdx.x;
  const int t = threadIdx.x;
  const float* htb = ht + (size_t)b * 2048 * 64;
  for (int i = t; i < 2048; i += 128) xs[i] = xx[b * 2048 + i];
  __syncthreads();

  const int wave = t >> 5;
  const int lane = t & 31;
  const int half = lane >> 4;
  const int m    = lane & 15;
  const int qA   = blockIdx.y * 128 + wave * 32;  // first query tile
  const int qB   = qA + 16;                       // second query tile

  // query operands (B matrices), fixed for the whole candidate sweep
  v2f bqA[16], bqB[16];
  {
    const v2f* pA = (const v2f*)(htb + (size_t)(qA + m) * 64);
    const v2f* pB = (const v2f*)(htb + (size_t)(qB + m) * 64);
#pragma unroll
    for (int kk = 0; kk < 16; ++kk) {
      bqA[kk] = pA[2 * kk + half];
      bqB[kk] = pB[2 * kk + half];
    }
  }

  Top4 tA, tB;
  tA.init(); tB.init();

  for (int mt = 0; mt < 128; ++mt) {
    const int m0 = mt * 16;
    const v2f* cp = (const v2f*)(htb + (size_t)(m0 + m) * 64);
    if (mt + 1 < 128)
      __builtin_prefetch(htb + (size_t)(m0 + 16 + m) * 64 + 32 * half, 0, 1);

    v2f ac[16];
#pragma unroll
    for (int kk = 0; kk < 16; ++kk) ac[kk] = cp[2 * kk + half];

    v8f accA = {0.f,0.f,0.f,0.f,0.f,0.f,0.f,0.f};
    v8f accB = {0.f,0.f,0.f,0.f,0.f,0.f,0.f,0.f};
#pragma unroll
    for (int kk = 0; kk < 16; ++kk) {
      accA = wmma_f32(ac[kk], bqA[kk], accA);
      accB = wmma_f32(ac[kk], bqB[kk], accB);
    }

#pragma unroll
    for (int v = 0; v < 8; ++v) {
      int ci = m0 + 8 * half + v;
      float xc = 0.5f * xs[ci];
      tA.insert(accA[v] - xc, ci);
      tB.insert(accB[v] - xc, ci);
    }
  }

  tA.merge_partner(lane);
  tB.merge_partner(lane);
  if (lane < 16) {
    int* oA = nidx + ((size_t)b * 2048 + qA + m) * 4;
    oA[0] = tA.i0; oA[1] = tA.i1; oA[2] = tA.i2; oA[3] = tA.i3;
    int* oB = nidx + ((size_t)b * 2048 + qB + m) * 4;
    oB[0] = tB.i0; oB[1] = tB.i1; oB[2] = tB.i2; oB[3] = tB.i3;
  }
}

// ---------------------------------------------------------------------------
// K4: edge conv. One wave = 16 pixels (4 points x 4 neighbors).
// conv2a (64x128) on WMMA f32; conv2b/c (64x64) on VALU through LDS;
// then max over the 4 neighbors -> lt[n][64] (transposed local).
// feat[c] = (c<64) ? h[c][nb]-h[c][ctr] : h[c-64][ctr]
// ---------------------------------------------------------------------------
__global__ __launch_bounds__(128) void k_edge(
    const float* __restrict__ ht, const int* __restrict__ nidx,
    const float* __restrict__ W2a, const float* __restrict__ s2a, const float* __restrict__ b2a,
    const float* __restrict__ W2b, const float* __restrict__ s2b, const float* __restrict__ b2b,
    const float* __restrict__ W2c, const float* __restrict__ s2c, const float* __restrict__ b2c,
    float* __restrict__ lt) {
  __shared__ float h2s[4][16][65];
  const int b    = blockIdx.x;
  const int t    = threadIdx.x;
  const int wave = t >> 5;
  const int lane = t & 31;
  const int half = lane >> 4;
  const int m    = lane & 15;       // pixel (and A-row) index
  const int pt   = m >> 2;
  const int j    = m & 3;
  const int n    = blockIdx.y * 16 + wave * 4 + pt;
  const int nb   = nidx[((size_t)b * 2048 + n) * 4 + j];
  const float* htb = ht + (size_t)b * 2048 * 64;

  // B operand: feat columns, contiguous v2f fetches from ht
  v2f fB[32];
  {
    const v2f* cn = (const v2f*)(htb + (size_t)n * 64);
    const v2f* cb = (const v2f*)(htb + (size_t)nb * 64);
#pragma unroll
    for (int kk = 0; kk < 16; ++kk) fB[kk] = cb[2 * kk + half] - cn[2 * kk + half];
#pragma unroll
    for (int kk = 16; kk < 32; ++kk) fB[kk] = cn[2 * (kk - 16) + half];
  }

  // conv2a via WMMA, 4 M-tiles of 16 out channels
  for (int tM = 0; tM < 4; ++tM) {
    const int o0 = tM * 16;
    v8f acc = {0.f,0.f,0.f,0.f,0.f,0.f,0.f,0.f};
#pragma unroll
    for (int kk = 0; kk < 32; ++kk) {
      const v2f* wp = (const v2f*)(W2a + (size_t)(o0 + m) * 128);
      acc = wmma_f32(wp[2 * kk + half], fB[kk], acc);
    }
#pragma unroll
    for (int v = 0; v < 8; ++v) {
      int o = o0 + 8 * half + v;
      float r = acc[v] * s2a[o] + b2a[o];
      h2s[wave][m][o] = r > 0.f ? r : 0.f;
    }
  }
  __syncthreads();

  // conv2b: lane computes 32 out channels for its pixel m
  float r2[32];
#pragma unroll 4
  for (int o = 0; o < 32; ++o) {
    int oo = 32 * half + o;
    float acc = 0.f;
    const float* w = W2b + oo * 64;
    for (int c = 0; c < 64; ++c) acc = fmaf(w[c], h2s[wave][m][c], acc);
    acc = acc * s2b[oo] + b2b[oo];
    r2[o] = acc > 0.f ? acc : 0.f;
  }
  __syncthreads();
#pragma unroll
  for (int o = 0; o < 32; ++o) h2s[wave][m][32 * half + o] = r2[o];
  __syncthreads();

  // conv2c
#pragma unroll 4
  for (int o = 0; o < 32; ++o) {
    int oo = 32 * half + o;
    float acc = 0.f;
    const float* w = W2c + oo * 64;
    for (int c = 0; c < 64; ++c) acc = fmaf(w[c], h2s[wave][m][c], acc);
    acc = acc * s2c[oo] + b2c[oo];
    r2[o] = acc > 0.f ? acc : 0.f;
  }
  __syncthreads();
#pragma unroll
  for (int o = 0; o < 32; ++o) h2s[wave][m][32 * half + o] = r2[o];
  __syncthreads();

  // max over the 4 neighbors -> lt[b][n][64]
  float* lb = lt + (size_t)b * 2048 * 64;
  for (int i = 0; i < 8; ++i) {
    int q = lane * 8 + i;
    int p = q >> 6;
    int o = q & 63;
    float mx = h2s[wave][p * 4 + 0][o];
    mx = fmaxf(mx, h2s[wave][p * 4 + 1][o]);
    mx = fmaxf(mx, h2s[wave][p * 4 + 2][o]);
    mx = fmaxf(mx, h2s[wave][p * 4 + 3][o]);
    lb[(size_t)(blockIdx.y * 16 + wave * 4 + p) * 64 + o] = mx;
  }
}

// ---------------------------------------------------------------------------
// K5: head conv W4a (256x192) on WMMA f32. final = concat(local64, gfin128).
// Writes h4 transposed: h4t[b][n][256] (contiguous 8-float rows per lane).
// ---------------------------------------------------------------------------
__global__ __launch_bounds__(128) void k_head(
    const float* __restrict__ lt, const float* __restrict__ gfin,
    const float* __restrict__ W4a, const float* __restrict__ b4a,
    float* __restrict__ h4t) {
  const int b    = blockIdx.x;
  const int n0   = blockIdx.y * 16;
  const int t    = threadIdx.x;
  const int wave = t >> 5;
  const int lane = t & 31;
  const int half = lane >> 4;
  const int m    = lane & 15;
  const float* lb = lt + (size_t)b * 2048 * 64;
  const float* gb = gfin + b * 128;

  v2f fB[48];
  {
    const v2f* lp = (const v2f*)(lb + (size_t)(n0 + m) * 64);
    const v2f* gp = (const v2f*)gb;
#pragma unroll
    for (int kk = 0; kk < 16; ++kk) fB[kk] = lp[2 * kk + half];
#pragma unroll
    for (int kk = 16; kk < 48; ++kk) fB[kk] = gp[2 * (kk - 16) + half];
  }

  float* h4b = h4t + (size_t)b * 2048 * 256;
  for (int tM = 0; tM < 4; ++tM) {
    const int o0 = (wave * 4 + tM) * 16;
    v8f acc = {0.f,0.f,0.f,0.f,0.f,0.f,0.f,0.f};
#pragma unroll
    for (int kk = 0; kk < 48; ++kk) {
      const v2f* wp = (const v2f*)(W4a + (size_t)(o0 + m) * 192);
      acc = wmma_f32(wp[2 * kk + half], fB[kk], acc);
    }
    float* dst = h4b + (size_t)(n0 + m) * 256 + o0 + 8 * half;
#pragma unroll
    for (int v = 0; v < 8; ++v) {
      float r = acc[v] + b4a[o0 + 8 * half + v];
      dst[v] = r > 0.f ? r : 0.f;
    }
  }
}

// ---------------------------------------------------------------------------
// K6: out = W4b(3x256) h4 + bias4b. Contiguous float4 reads of h4t rows.
// ---------------------------------------------------------------------------
__global__ __launch_bounds__(256) void k_out(
    const float* __restrict__ h4t,
    const float* __restrict__ W4b, const float* __restrict__ b4b,
    float* __restrict__ out) {
  int gid = blockIdx.x * 256 + threadIdx.x;
  if (gid >= NBATCH * NPTS) return;
  int b = gid >> 11;
  int n = gid & 2047;
  const float4* hc = (const float4*)(h4t + ((size_t)b * 2048 + n) * 256);
  float a0 = b4b[0], a1 = b4b[1], a2 = b4b[2];
#pragma unroll 4
  for (int c4 = 0; c4 < 64; ++c4) {
    float4 v = hc[c4];
    int c = c4 * 4;
    a0 = fmaf(W4b[c + 0], v.x, a0); a0 = fmaf(W4b[c + 1], v.y, a0);
    a0 = fmaf(W4b[c + 2], v.z, a0); a0 = fmaf(W4b[c + 3], v.w, a0);
    a1 = fmaf(W4b[256 + c + 0], v.x, a1); a1 = fmaf(W4b[256 + c + 1], v.y, a1);
    a1 = fmaf(W4b[256 + c + 2], v.z, a1); a1 = fmaf(W4b[256 + c + 3], v.w, a1);
    a2 = fmaf(W4b[512 + c + 0], v.x, a2); a2 = fmaf(W4b[512 + c + 1], v.y, a2);
    a2 = fmaf(W4b[512 + c + 2], v.z, a2); a2 = fmaf(W4b[512 + c + 3], v.w, a2);
  }
  float* ob = out + (size_t)b * 3 * 2048;
  ob[n] = a0; ob[2048 + n] = a1; ob[2 * 2048 + n] = a2;
}

// ---------------------------------------------------------------------------
extern "C" void kernel_launch(void* const* d_in, const int* in_sizes, int n_in,
                              void* d_out, int out_size, void* d_ws, size_t ws_size,
                              hipStream_t stream) {
  const float* x     = (const float*)d_in[0];
  const float* W_dup = (const float*)d_in[1];
  const float* s_dup = (const float*)d_in[2];
  const float* b_dup = (const float*)d_in[3];
  const float* W_c1  = (const float*)d_in[4];
  const float* s_c1  = (const float*)d_in[5];
  const float* b_c1  = (const float*)d_in[6];
  const float* W2a = (const float*)d_in[7];
  const float* s2a = (const float*)d_in[8];
  const float* b2a = (const float*)d_in[9];
  const float* W2b = (const float*)d_in[10];
  const float* s2b = (const float*)d_in[11];
  const float* b2b = (const float*)d_in[12];
  const float* W2c = (const float*)d_in[13];
  const float* s2c = (const float*)d_in[14];
  const float* b2c = (const float*)d_in[15];
  const float* W3a = (const float*)d_in[16];
  const float* s3a = (const float*)d_in[17];
  const float* b3a = (const float*)d_in[18];
  const float* W3b = (const float*)d_in[19];
  const float* s3b = (const float*)d_in[20];
  const float* b3b = (const float*)d_in[21];
  const float* W4a    = (const float*)d_in[22];
  const float* bias4a = (const float*)d_in[23];
  const float* W4b    = (const float*)d_in[24];
  const float* bias4b = (const float*)d_in[25];
  float* out = (float*)d_out;

  // workspace carve-up (~29.6 MB)
  float* h     = (float*)d_ws;                          // 8*64*2048
  float* ht    = h  + (size_t)8 * 64 * 2048;            // 8*2048*64
  float* xx    = ht + (size_t)8 * 2048 * 64;            // 8*2048
  float* gfin  = xx + (size_t)8 * 2048;                 // 8*128
  int*   nidx  = (int*)(gfin + (size_t)8 * 128);        // 8*2048*4
  float* lt    = (float*)(nidx + (size_t)8 * 2048 * 4); // 8*2048*64
  float* h4t   = lt + (size_t)8 * 2048 * 64;            // 8*2048*256

  k_front<<<dim3(8, 32), 128, 0, stream>>>(x, W_dup, s_dup, b_dup, W_c1, s_c1, b_c1, h, ht);
  k_stats<<<dim3(8), 256, 0, stream>>>(h, ht, W3a, s3a, b3a, W3b, s3b, b3b, xx, gfin);
  k_knn<<<dim3(8, 16), 128, 0, stream>>>(ht, xx, nidx);
  k_edge<<<dim3(8, 128), 128, 0, stream>>>(ht, nidx, W2a, s2a, b2a, W2b, s2b, b2b,
                                           W2c, s2c, b2c, lt);
  k_head<<<dim3(8, 128), 128, 0, stream>>>(lt, gfin, W4a, bias4a, h4t);
  k_out<<<dim3(64), 256, 0, stream>>>(h4t, W4b, bias4b, out);
}